// AttentionMechanism_62835371540891
// MI455X (gfx1250) — compile-verified
//
#include <hip/hip_runtime.h>
#include <hip/hip_bf16.h>
#include <math.h>

// Q (B,D)=(1024,128) f32, K/V (B,H,D)=(1024,2048,128) f32,
// mask (B,H) bool (1B/elem, nonzero => masked out). Output (B,D) f32.
//
// One block per batch row b; 256 threads = 8 wave32s.
// Pass A: scores via V_WMMA_F32_16X16X4_F32, branchless mask handling
//         (masked lanes read a dummy cached row; garbage scores discarded).
// Pass B: masked stable softmax over H.
// Pass C: context via lane-wide float4 V streaming, skipping zero-weight rows.

typedef __attribute__((ext_vector_type(2))) float v2f;
typedef __attribute__((ext_vector_type(8))) float v8f;

#define THREADS 256
#define WAVES   8
#define H_MAX   2048
#define D_FIX   128

__global__ __launch_bounds__(THREADS)
void attn_masked_softmax_kernel(const float* __restrict__ Q,
                                const float* __restrict__ K,
                                const float* __restrict__ V,
                                const unsigned char* __restrict__ mask,
                                float* __restrict__ out,
                                int B, int H, int D)
{
    const int b    = blockIdx.x;
    const int tid  = threadIdx.x;
    const int lane = tid & 31;
    const int wave = tid >> 5;

    __shared__ __align__(16) float Qs[D_FIX];       // Q[b,:]
    __shared__ float  scores_s[H_MAX];              // scores -> masked -> exp weights
    __shared__ float  red[WAVES];
    __shared__ float  bcast[2];
    __shared__ float4 partial[WAVES][32];

    // ---- stage Q[b] into LDS ----
    if (tid < 32) {
        ((float4*)Qs)[tid] = ((const float4*)(Q + (size_t)b * D))[tid];
    }
    __syncthreads();

    // ---------------- Pass A: scores via WMMA f32 16x16x4 ----------------
    // A 16x4 layout: lanes 0-15 -> M=lane, K=0,1 in vgpr0,1; lanes 16-31 -> M=lane-16, K=2,3.
    // B 4x16 layout: rows 0,2 in vgpr0 (lane halves), rows 1,3 in vgpr1; Q only in column n=0.
    const int  m    = lane & 15;
    const int  koff = (lane >> 4) * 2;     // 0 (lanes 0-15) / 2 (lanes 16-31)
    const bool col0 = (m == 0);            // lanes 0 and 16 carry B column 0

    const float* kbatch = K + (size_t)b * H * (size_t)D;   // dummy row for masked lanes
    const float* qlane  = Qs + koff;

    const int tiles = H >> 4;
    for (int t = wave; t < tiles; t += WAVES) {
        const int h0 = t << 4;
        // Branchless mask skip: masked lanes point at row 0 of K[b] (L0-resident,
        // ~zero extra HBM traffic); their scores are garbage but pass B overwrites
        // masked entries with -inf before the softmax. WMMA rows are independent,
        // so unmasked rows are unaffected. This halves K traffic on average.
        const bool  active = (mask[(size_t)b * H + h0 + m] == 0);
        const float* rowp  = active ? (kbatch + (size_t)(h0 + m) * D + koff) : (kbatch + koff);

        v8f c = {};
        #pragma unroll 8
        for (int kk = 0; kk < D_FIX; kk += 4) {
            const v2f a = *(const v2f*)(rowp + kk);        // unconditional b64 load
            const float q0 = qlane[kk];
            const float q1 = qlane[kk + 1];
            v2f bf;
            bf.x = col0 ? q0 : 0.0f;
            bf.y = col0 ? q1 : 0.0f;
            c = __builtin_amdgcn_wmma_f32_16x16x4_f32(
                    false, a, false, bf, (short)0, c, false, false);
        }
        // D layout: vgpr r, lane 0 -> D[m=r, n=0]; lane 16 -> D[m=8+r, n=0]
        if (lane == 0) {
            #pragma unroll
            for (int r = 0; r < 8; ++r) scores_s[h0 + r] = c[r];
        } else if (lane == 16) {
            #pragma unroll
            for (int r = 0; r < 8; ++r) scores_s[h0 + 8 + r] = c[r];
        }
    }
    __syncthreads();

    // ---------------- Pass B: masked stable softmax over H ----------------
    const unsigned char* mrow = mask + (size_t)b * H;

    float lmax = -__builtin_inff();
    for (int h = tid; h < H; h += THREADS) {
        const float s = mrow[h] ? -__builtin_inff() : scores_s[h];
        scores_s[h] = s;
        lmax = fmaxf(lmax, s);
    }
    #pragma unroll
    for (int off = 16; off > 0; off >>= 1)
        lmax = fmaxf(lmax, __shfl_xor(lmax, off, 32));
    if (lane == 0) red[wave] = lmax;
    __syncthreads();
    if (tid == 0) {
        float g = red[0];
        #pragma unroll
        for (int w = 1; w < WAVES; ++w) g = fmaxf(g, red[w]);
        bcast[0] = g;
    }
    __syncthreads();
    const float gmax      = bcast[0];
    const bool  allmasked = !(gmax > -__builtin_inff());

    float lsum = 0.0f;
    if (!allmasked) {
        for (int h = tid; h < H; h += THREADS) {
            const float e = __expf(scores_s[h] - gmax);    // masked: exp(-inf)=0 exactly
            scores_s[h] = e;
            lsum += e;
        }
    }
    #pragma unroll
    for (int off = 16; off > 0; off >>= 1)
        lsum += __shfl_xor(lsum, off, 32);
    if (lane == 0) red[wave] = lsum;
    __syncthreads();
    if (tid == 0) {
        float s = 0.0f;
        #pragma unroll
        for (int w = 0; w < WAVES; ++w) s += red[w];
        bcast[1] = s;
    }
    __syncthreads();
    const float inv = allmasked ? 0.0f : (1.0f / bcast[1]);

    // ---------------- Pass C: context = (e/sum)^T V[b] ----------------
    float4 acc = make_float4(0.0f, 0.0f, 0.0f, 0.0f);
    if (!allmasked) {
        const float* vbase = V + (size_t)b * H * (size_t)D;
        for (int h = wave; h < H; h += WAVES) {
            const float w = scores_s[h];           // wave-uniform
            if (w == 0.0f) continue;               // skip masked/underflowed rows
            const float4 vv = *(const float4*)(vbase + (size_t)h * D + lane * 4);
            acc.x += w * vv.x;
            acc.y += w * vv.y;
            acc.z += w * vv.z;
            acc.w += w * vv.w;
        }
    }
    partial[wave][lane] = acc;
    __syncthreads();

    if (wave == 0) {
        float4 s = partial[0][lane];
        #pragma unroll
        for (int w = 1; w < WAVES; ++w) {
            const float4 p = partial[w][lane];
            s.x += p.x; s.y += p.y; s.z += p.z; s.w += p.w;
        }
        s.x *= inv; s.y *= inv; s.z *= inv; s.w *= inv;
        ((float4*)out)[(size_t)b * (D >> 2) + lane] = s;
    }
}

extern "C" void kernel_launch(void* const* d_in, const int* in_sizes, int n_in,
                              void* d_out, int out_size, void* d_ws, size_t ws_size,
                              hipStream_t stream) {
    const float*         Q    = (const float*)d_in[0];
    const float*         K    = (const float*)d_in[1];
    const float*         V    = (const float*)d_in[2];
    const unsigned char* mask = (const unsigned char*)d_in[3];
    float*               out  = (float*)d_out;

    const int BD  = in_sizes[0];   // B*D
    const int BHD = in_sizes[1];   // B*H*D
    const int BH  = in_sizes[3];   // B*H
    const int D   = BHD / BH;      // 128
    const int B   = BD / D;        // 1024
    const int H   = BH / B;        // 2048

    attn_masked_softmax_kernel<<<B, THREADS, 0, stream>>>(Q, K, V, mask, out, B, H, D);
}